// MotionBertTokenizer_68092411510796
// MI455X (gfx1250) — compile-verified
//
#include <hip/hip_runtime.h>
#include <stdint.h>

typedef __attribute__((ext_vector_type(2))) float        v2f;
typedef __attribute__((ext_vector_type(4))) float        v4f;
typedef __attribute__((ext_vector_type(8))) float        v8f;
typedef __attribute__((ext_vector_type(4))) unsigned int v4u;
typedef __attribute__((ext_vector_type(4))) int          v4i;
typedef __attribute__((ext_vector_type(8))) int          v8i;

#define T_LEN   768
#define NJ      17
#define ROW     51                      // 17 joints * 3 comps
#define MAXAGE  512
#define OUT_PER_BN (MAXAGE * ROW)       // 26112 floats = 104448 B
#define IN_PER_BN  (T_LEN * ROW)        // 39168 floats
#define NWAVES  8
#define ITERS   (T_LEN / 16 / NWAVES)   // 6 token-groups of 16 per wave
#define GRP_F   (16 * ROW)              // 816 floats per group
#define GRP_V4  (GRP_F / 4)             // 204 float4 per group (16B aligned)

// Shared tile: outBuf MUST be first member (TDM descriptor assumes LDS offset 0).
struct SMem {
    float outBuf[OUT_PER_BN];           // 104448 B  per-(b,n) output tile
    float stage[NWAVES][GRP_F];         //  26112 B  per-wave 16-token staging
    int   slot[NWAVES][16];             //    512 B  scatter slot per token (-1 invalid)
};

// posetrack -> h36m remap matrix entries (rows 0..15; row 16 handled scalar)
__device__ __forceinline__ float M_entry(int i, int j) {
    switch (i) {
        case 0:  return (j == 11 || j == 12) ? 0.5f : 0.0f;
        case 1:  return (j == 12) ? 1.0f : 0.0f;
        case 2:  return (j == 14) ? 1.0f : 0.0f;
        case 3:  return (j == 16) ? 1.0f : 0.0f;
        case 4:  return (j == 11) ? 1.0f : 0.0f;
        case 5:  return (j == 13) ? 1.0f : 0.0f;
        case 6:  return (j == 15) ? 1.0f : 0.0f;
        case 7:  return (j == 5 || j == 6 || j == 11 || j == 12) ? 0.25f : 0.0f;
        case 8:  return (j == 5 || j == 6) ? 0.5f : 0.0f;
        case 9:  return (j == 1) ? 1.0f : 0.0f;
        case 10: return (j == 2) ? 1.0f : 0.0f;
        case 11: return (j == 5) ? 1.0f : 0.0f;
        case 12: return (j == 7) ? 1.0f : 0.0f;
        case 13: return (j == 9) ? 1.0f : 0.0f;
        case 14: return (j == 6) ? 1.0f : 0.0f;
        case 15: return (j == 8) ? 1.0f : 0.0f;
        default: return 0.0f;
    }
}

__global__ __launch_bounds__(256)
void mbt_kernel(const float* __restrict__ kp,
                const float* __restrict__ ages,
                float* __restrict__ out)
{
    __shared__ SMem sm;
    const int tid  = threadIdx.x;
    const int wv   = tid >> 5;
    const int lane = tid & 31;
    const int hi   = lane >> 4;          // lane half (WMMA K/M split)
    const int lo   = lane & 15;
    const int bn   = blockIdx.x;         // one workgroup per (b,n)

    // Zero the per-(b,n) output tile in LDS with B128 stores (untouched slots stay 0).
    {
        v4f z = {0.f, 0.f, 0.f, 0.f};
        v4f* ob4 = (v4f*)sm.outBuf;
        for (int i = tid; i < OUT_PER_BN / 4; i += 256) ob4[i] = z;
    }

    // Preload A fragments: 16x4 f32 tiles of M for 5 K-chunks (K padded 17->20).
    // Layout: VGPR0 lanes0-15 K=k0, lanes16-31 K=k0+2; VGPR1 K=k0+1 / k0+3.
    v2f Afrag[5];
#pragma unroll
    for (int c = 0; c < 5; ++c) {
        const int k = 4 * c + 2 * hi;
        Afrag[c].x = M_entry(lo, k);
        Afrag[c].y = M_entry(lo, k + 1);
    }

    const float* kp_bn  = kp   + (size_t)bn * IN_PER_BN;
    const float* age_bn = ages + (size_t)bn * T_LEN;
    float* dst = sm.stage[wv];

    __syncthreads();   // tile fully zeroed before any wave scatters into it

    for (int it = 0; it < ITERS; ++it) {
        const int tbase = (wv + NWAVES * it) * 16;   // base token of this group

        // Stage 16 contiguous tokens: 816 floats = 204 x B128, 16B aligned.
        const v4f* src4 = (const v4f*)(kp_bn + (size_t)tbase * ROW);
        v4f* dst4 = (v4f*)dst;
        for (int i = lane; i < GRP_V4; i += 32) dst4[i] = src4[i];

        // Validity + flipped scatter slot per token.
        if (lane < 16) {
            const float a = age_bn[tbase + lane];
            int s = -1;
            if (a == a && a >= 0.0f && a < (float)MAXAGE)  // finite, in range
                s = (MAXAGE - 1) - (int)a;                 // flip along age axis
            sm.slot[wv][lane] = s;
        }
        // stage/slot are wave-private; same-wave LDS ops are in-order (no block barrier).
        __builtin_amdgcn_wave_barrier();

        // Joint remap via WMMA f32 16x16x4: D[joint 0..15][token 0..15], per channel.
        v8f acc[3];
#pragma unroll
        for (int ch = 0; ch < 3; ++ch) {
            v8f cacc = {0.f, 0.f, 0.f, 0.f, 0.f, 0.f, 0.f, 0.f};
#pragma unroll
            for (int kc = 0; kc < 5; ++kc) {
                const int k0 = 4 * kc + 2 * hi;
                v2f b;
                b.x = (k0     < NJ) ? dst[lo * ROW + k0 * 3 + ch]       : 0.0f;
                b.y = (k0 + 1 < NJ) ? dst[lo * ROW + (k0 + 1) * 3 + ch] : 0.0f;
                cacc = __builtin_amdgcn_wmma_f32_16x16x4_f32(
                    /*neg_a=*/false, Afrag[kc], /*neg_b=*/false, b,
                    /*c_mod=*/(short)0, cacc, /*reuse_a=*/false, /*reuse_b=*/false);
            }
            acc[ch] = cacc;
        }

        // Scatter joints 0..15 of this lane's token into the LDS tile.
        // D layout: VGPR r -> joint r (lanes 0-15) / joint r+8 (lanes 16-31), token = lo.
        const int slot = sm.slot[wv][lo];
        if (slot >= 0) {
            float* ob = sm.outBuf + slot * ROW;
#pragma unroll
            for (int r = 0; r < 8; ++r) {
                const int j  = r + 8 * hi;
                const float cf = acc[2][r];
                const float x  = (cf == 0.0f) ? 0.0f : (2.0f * acc[0][r] - 1.0f);
                const float y  = (cf == 0.0f) ? 0.0f : (2.0f * acc[1][r] - 1.0f);
                ob[j * 3 + 0] = x;
                ob[j * 3 + 1] = y;
                ob[j * 3 + 2] = cf;
            }
        }
        // h36m joint 16 = posetrack joint 10 (scalar copy path).
        if (lane < 16) {
            const int s = sm.slot[wv][lane];
            if (s >= 0) {
                const float cf = dst[lane * ROW + 32];   // conf of joint 10
                const float x  = (cf == 0.0f) ? 0.0f : (2.0f * dst[lane * ROW + 30] - 1.0f);
                const float y  = (cf == 0.0f) ? 0.0f : (2.0f * dst[lane * ROW + 31] - 1.0f);
                float* ob = sm.outBuf + s * ROW + 48;
                ob[0] = x; ob[1] = y; ob[2] = cf;
            }
        }
        __builtin_amdgcn_wave_barrier();   // keep restage of wave-private buffer ordered
    }

    __syncthreads();   // all waves' scatters into outBuf complete

    // Stream the finished 104448-byte tile to global via the Tensor Data Mover.
    if (tid < 32) {   // wave 0 issues the TDM op (wave-level, EXEC-independent)
        const unsigned long long gaddr =
            (unsigned long long)(uintptr_t)(out + (size_t)bn * OUT_PER_BN);
        // D# group 0: count=1 | lds_addr=0 (outBuf is first LDS object) | global_addr | type=2
        v4u g0;
        g0.x = 1u;
        g0.y = 0u;
        g0.z = (unsigned int)(gaddr & 0xFFFFFFFFull);
        g0.w = (unsigned int)((gaddr >> 32) & 0x01FFFFFFull) | (2u << 30);
        // D# group 1: data_size=4B, 1D tile: tensor_dim0=tile_dim0=stride=26112, dim1=1
        v8i g1;
        g1[0] = 0x00020000;                         // wg_mask=0, data_size=2 (4 bytes)
        g1[1] = (int)((OUT_PER_BN & 0xFFFF) << 16); // tensor_dim0[15:0] @ bits 63:48
        g1[2] = 0x00010000;                         // tensor_dim0 hi=0; tensor_dim1=1
        g1[3] = (int)((OUT_PER_BN & 0xFFFF) << 16); // tile_dim0 @ bits 127:112
        g1[4] = 1;                                  // tile_dim1=1, tile_dim2=0
        g1[5] = OUT_PER_BN;                         // tensor_dim0_stride[31:0]
        g1[6] = 0;
        g1[7] = 0;
        v4i gz = {0, 0, 0, 0};
#if defined(__clang_major__) && (__clang_major__ >= 23)
        v8i gz8 = {0, 0, 0, 0, 0, 0, 0, 0};
        __builtin_amdgcn_tensor_store_from_lds(g0, g1, gz, gz, gz8, 0);
#else
        __builtin_amdgcn_tensor_store_from_lds(g0, g1, gz, gz, 0);
#endif
        __builtin_amdgcn_s_wait_tensorcnt(0);
    }
}

extern "C" void kernel_launch(void* const* d_in, const int* in_sizes, int n_in,
                              void* d_out, int out_size, void* d_ws, size_t ws_size,
                              hipStream_t stream) {
    const float* kp   = (const float*)d_in[0];   // [B,N,T,17,3] f32
    const float* ages = (const float*)d_in[1];   // [B,N,T,1]   f32
    float* out = (float*)d_out;                  // [B,N,512,17,3] f32
    const int bn = in_sizes[0] / IN_PER_BN;      // B*N = 1024
    mbt_kernel<<<bn, 256, 0, stream>>>(kp, ages, out);
    (void)d_ws; (void)ws_size; (void)n_in; (void)out_size;
}